// Attention_28561532519042
// MI455X (gfx1250) — compile-verified
//
#include <hip/hip_runtime.h>
#include <hip/hip_bf16.h>
#include <math.h>

// ---------------- problem dims ----------------
#define B_    16
#define V_    25
#define B2_   (B_ * V_)      // 400
#define T_    120
#define C_    128
#define NHA   3
#define QKV_C 1536
#define HEADS 8
#define DH    64
#define INNER 512
#define KCAT  512            // 3*128 (A_h-mixed) + 128 (down path)
#define TB    4              // time-steps per gcn_qkv block (weight reuse x4)
#define EPSF  1e-5f

typedef float v2f __attribute__((ext_vector_type(2)));
typedef float v8f __attribute__((ext_vector_type(8)));

// D(16x16,f32) = A(16x4,f32) x B(4x16,f32) + C  -- full-precision CDNA5 WMMA
__device__ __forceinline__ v8f wmma_f32(v2f a, v2f b, v8f c) {
  return __builtin_amdgcn_wmma_f32_16x16x4_f32(false, a, false, b, (short)0, c,
                                               false, false);
}

// ---------------------------------------------------------------------------
// Kernel 0: fold BN (eval) into concatenated weights + single bias vector.
// ---------------------------------------------------------------------------
__global__ void fold_weights(const float* __restrict__ conv_w,
                             const float* __restrict__ conv_b,
                             const float* __restrict__ bng, const float* __restrict__ bnb,
                             const float* __restrict__ bnm, const float* __restrict__ bnv,
                             const float* __restrict__ down_w, const float* __restrict__ down_b,
                             const float* __restrict__ dbg, const float* __restrict__ dbb,
                             const float* __restrict__ dbm, const float* __restrict__ dbv,
                             float* __restrict__ Wfold, float* __restrict__ bias) {
  int idx = blockIdx.x * blockDim.x + threadIdx.x;
  const int total = KCAT * QKV_C;
  if (idx < total) {
    int k  = idx / QKV_C;
    int oc = idx - k * QKV_C;
    float w;
    if (k < NHA * C_) {
      int h = k >> 7;
      int c = k & (C_ - 1);
      float inv = bng[oc] * rsqrtf(bnv[oc] + EPSF);
      w = inv * conv_w[(h * QKV_C + oc) * C_ + c];
    } else {
      int c = k - NHA * C_;
      float inv = dbg[oc] * rsqrtf(dbv[oc] + EPSF);
      w = inv * down_w[oc * C_ + c];
    }
    Wfold[k * QKV_C + oc] = w;
  }
  if (idx < QKV_C) {
    int oc = idx;
    float inv  = bng[oc] * rsqrtf(bnv[oc] + EPSF);
    float dinv = dbg[oc] * rsqrtf(dbv[oc] + EPSF);
    float bsum = conv_b[oc] + conv_b[QKV_C + oc] + conv_b[2 * QKV_C + oc];
    bias[oc] = inv * bsum + (bnb[oc] - bnm[oc] * inv)
             + dinv * down_b[oc] + (dbb[oc] - dbm[oc] * dinv);
  }
}

// Seed out with out_b (heads accumulate with atomics afterwards)
__global__ void init_out(const float* __restrict__ out_b, float* __restrict__ out) {
  int idx = blockIdx.x * blockDim.x + threadIdx.x;
  if (idx < B2_ * T_ * C_) out[idx] = out_b[idx & (C_ - 1)];
}

// ---------------------------------------------------------------------------
// Kernel 1: per (b, t0..t0+3): build Kcat = [A0@X; A1@X; A2@X; X] per step
// (32x512, zero-padded rows 25..31) stacked into a 128x512 LDS block, then
// qkv = relu(Kcat @ Wfold + bias) via f32 WMMA.
// 272KB LDS -> 1 workgroup/WGP, so launch_bounds frees the full VGPR budget;
// per k-step: 8 A-frags loaded into distinct regs, then 8 WMMAs (pipelined).
// ---------------------------------------------------------------------------
__global__ void __launch_bounds__(256, 1)
gcn_qkv(const float* __restrict__ x, const float* __restrict__ Aadj,
        const float* __restrict__ Wfold, const float* __restrict__ bias,
        float* __restrict__ qkv) {
  extern __shared__ float smem[];
  const int SKS = KCAT + 4;              // 516-float row stride (bank-skewed)
  float* sK = smem;                      // [128][516]  (TB*32 rows)
  float* sA = smem + (TB * 32) * SKS;    // [3][25][25]

  const int b  = blockIdx.x / (T_ / TB);
  const int t0 = (blockIdx.x % (T_ / TB)) * TB;
  const int tid = threadIdx.x;

  for (int i = tid; i < TB * 32 * SKS; i += 256) sK[i] = 0.f;
  for (int i = tid; i < NHA * V_ * V_; i += 256) sA[i] = Aadj[i];
  __syncthreads();

  // X rows into cols [384,512) of each 32-row group
  for (int i = tid; i < TB * V_ * C_; i += 256) {
    int tb = i / (V_ * C_);
    int rem = i - tb * V_ * C_;
    int u = rem >> 7, c = rem & (C_ - 1);
    sK[(tb * 32 + u) * SKS + NHA * C_ + c] =
        x[((b * V_ + u) * T_ + t0 + tb) * C_ + c];
  }
  __syncthreads();

  // xa_h = A_h @ X into cols [h*128, h*128+128)   (small: VALU is fine)
  for (int i = tid; i < TB * NHA * V_ * C_; i += 256) {
    int tb = i / (NHA * V_ * C_);
    int rem = i - tb * NHA * V_ * C_;
    int h = rem / (V_ * C_);
    rem -= h * V_ * C_;
    int v = rem >> 7, c = rem & (C_ - 1);
    const float* Ar = &sA[(h * V_ + v) * V_];
    float acc = 0.f;
#pragma unroll
    for (int u = 0; u < V_; ++u) acc += Ar[u] * sK[(tb * 32 + u) * SKS + NHA * C_ + c];
    sK[(tb * 32 + v) * SKS + h * C_ + c] = acc;
  }
  __syncthreads();

  const int wave = tid >> 5;
  const int lane = tid & 31;
  const int l  = lane & 15;
  const int hf = lane >> 4;

  for (int n = wave; n < QKV_C / 16; n += 8) {
    const int oc = n * 16 + l;
    const float bv = bias[oc];
    const float* Wcol = Wfold + oc;      // column oc, row stride QKV_C (L2-hot)
    v8f acc[2 * TB];
#pragma unroll
    for (int m = 0; m < 2 * TB; ++m) acc[m] = (v8f){};
    for (int k0 = 0; k0 < KCAT; k0 += 4) {
      int kk = k0 + 2 * hf;
      v2f bf; bf[0] = Wcol[(size_t)kk * QKV_C]; bf[1] = Wcol[(size_t)(kk + 1) * QKV_C];
      v2f a[2 * TB];
#pragma unroll
      for (int m = 0; m < 2 * TB; ++m) {       // 8 DS loads in flight
        a[m][0] = sK[(m * 16 + l) * SKS + kk];
        a[m][1] = sK[(m * 16 + l) * SKS + kk + 1];
      }
#pragma unroll
      for (int m = 0; m < 2 * TB; ++m)         // B reused across 8 M tiles
        acc[m] = wmma_f32(a[m], bf, acc[m]);
    }
#pragma unroll
    for (int m = 0; m < 2 * TB; ++m) {
      int t = t0 + (m >> 1);
#pragma unroll
      for (int r = 0; r < 8; ++r) {
        int vloc = (m & 1) * 16 + r + 8 * hf;  // D layout: M = r + 8*half
        float y = acc[m][r] + bv;
        y = y > 0.f ? y : 0.f;
        if (vloc < V_)
          qkv[(((size_t)(b * V_ + vloc)) * T_ + t) * QKV_C + oc] = y;
      }
    }
  }
}

// ---------------------------------------------------------------------------
// Kernel 2: per (b2, head): causal attention (T padded 120->128), fused with
// the out-projection (atomicAdd accumulation over heads).
// 8 waves; wave w owns the 16-row stripe [16w, 16w+16).
// ---------------------------------------------------------------------------
__global__ void __launch_bounds__(256, 1)
attention(const float* __restrict__ qkv, const float* __restrict__ out_w,
          float* __restrict__ out, float* __restrict__ attn) {
  extern __shared__ float smem[];
  const int SQ = 68, SP = 132;           // bank-skewed strides
  float* sQ  = smem;                     // [128][68]
  float* sKm = sQ  + 128 * SQ;           // [128][68]
  float* sV  = sKm + 128 * SQ;           // [128][68]
  float* sO  = sV  + 128 * SQ;           // [128][68]
  float* sP  = sO  + 128 * SQ;           // [128][132] softmax probs

  const int b2 = blockIdx.x;
  const int h  = blockIdx.y;
  const int tid = threadIdx.x;

  for (int i = tid; i < 128 * DH; i += 256) {
    int tr = i >> 6, d = i & (DH - 1);
    float q = 0.f, k = 0.f, vv = 0.f;
    if (tr < T_) {
      const float* base = qkv + ((size_t)b2 * T_ + tr) * QKV_C + h * DH + d;
      q = base[0]; k = base[INNER]; vv = base[2 * INNER];
    }
    sQ[tr * SQ + d] = q; sKm[tr * SQ + d] = k; sV[tr * SQ + d] = vv;
  }
  __syncthreads();

  const int wave = tid >> 5, lane = tid & 31;
  const int l  = lane & 15;
  const int hf = lane >> 4;
  const float scale = 0.125f;            // 64^-0.5
  const float NEG = -INFINITY;

  // ---- dots = scale * Q K^T, causal-masked ----
  // Q-fragments are n-invariant: load the 16 k-step frags once, reuse per tile.
  v2f aq[16];
#pragma unroll
  for (int d4 = 0; d4 < 16; ++d4) {
    int dd = d4 * 4 + 2 * hf;
    aq[d4][0] = sQ[(wave * 16 + l) * SQ + dd];
    aq[d4][1] = sQ[(wave * 16 + l) * SQ + dd + 1];
  }
  float P[8][8];
  for (int n = 0; n <= wave; ++n) {      // tiles n>wave are fully causal-masked
    v8f acc = {};
#pragma unroll
    for (int d4 = 0; d4 < 16; ++d4) {
      int dd = d4 * 4 + 2 * hf;
      v2f bb; bb[0] = sKm[(n * 16 + l) * SQ + dd];
              bb[1] = sKm[(n * 16 + l) * SQ + dd + 1];
      acc = wmma_f32(aq[d4], bb, acc);
    }
#pragma unroll
    for (int r = 0; r < 8; ++r) {
      int i = wave * 16 + r + 8 * hf;    // D layout: M = r + 8*half
      int j = n * 16 + l;                // N = lane % 16
      P[n][r] = (i < T_ && j <= i) ? acc[r] * scale : NEG;
    }
  }
  for (int n = wave + 1; n < 8; ++n)
#pragma unroll
    for (int r = 0; r < 8; ++r) P[n][r] = NEG;

  // ---- row softmax: reduce over n (regs) and 16 lanes (shfl, width 16) ----
#pragma unroll
  for (int r = 0; r < 8; ++r) {
    float m = NEG;
#pragma unroll
    for (int n = 0; n < 8; ++n) m = fmaxf(m, P[n][r]);
    for (int s = 1; s < 16; s <<= 1) m = fmaxf(m, __shfl_xor(m, s, 16));
    float sum = 0.f;
#pragma unroll
    for (int n = 0; n < 8; ++n) {
      float p = (P[n][r] == NEG) ? 0.f : __expf(P[n][r] - m);
      P[n][r] = p; sum += p;
    }
    for (int s = 1; s < 16; s <<= 1) sum += __shfl_xor(sum, s, 16);
    float inv = sum > 0.f ? 1.f / sum : 0.f;   // padded rows -> all-zero
#pragma unroll
    for (int n = 0; n < 8; ++n) P[n][r] *= inv;
  }

  // ---- emit attn + stage probs for P@V ----
#pragma unroll
  for (int r = 0; r < 8; ++r) {
    int i = wave * 16 + r + 8 * hf;
#pragma unroll
    for (int n = 0; n < 8; ++n) {
      float p = P[n][r];
      int j = n * 16 + l;
      sP[i * SP + j] = p;
      if (i < T_ && j < T_)
        attn[(((size_t)b2 * HEADS + h) * T_ + i) * T_ + j] = p;
    }
  }
  __syncthreads();

  // ---- O = P @ V (16x64 per wave, K=128); A-frag shared across 4 N tiles ----
  {
    v8f accv[4];
#pragma unroll
    for (int n = 0; n < 4; ++n) accv[n] = (v8f){};
    for (int k0 = 0; k0 < 128; k0 += 4) {
      int kk = k0 + 2 * hf;
      v2f a; a[0] = sP[(wave * 16 + l) * SP + kk];
             a[1] = sP[(wave * 16 + l) * SP + kk + 1];
#pragma unroll
      for (int n = 0; n < 4; ++n) {
        v2f bb; bb[0] = sV[kk * SQ + n * 16 + l];
                bb[1] = sV[(kk + 1) * SQ + n * 16 + l];
        accv[n] = wmma_f32(a, bb, accv[n]);
      }
    }
#pragma unroll
    for (int n = 0; n < 4; ++n)
#pragma unroll
      for (int r = 0; r < 8; ++r)
        sO[(wave * 16 + r + 8 * hf) * SQ + n * 16 + l] = accv[n][r];
  }
  __syncthreads();

  // ---- out += O @ W_h^T (L2-resident); A-frag shared across 8 N tiles ----
  {
    v8f acco[8];
#pragma unroll
    for (int n = 0; n < 8; ++n) acco[n] = (v8f){};
    for (int d0 = 0; d0 < DH; d0 += 4) {
      int dd = d0 + 2 * hf;
      v2f a; a[0] = sO[(wave * 16 + l) * SQ + dd];
             a[1] = sO[(wave * 16 + l) * SQ + dd + 1];
#pragma unroll
      for (int n = 0; n < 8; ++n) {
        const float* wrow = out_w + (size_t)(n * 16 + l) * INNER + h * DH;
        v2f bb; bb[0] = wrow[dd]; bb[1] = wrow[dd + 1];
        acco[n] = wmma_f32(a, bb, acco[n]);
      }
    }
#pragma unroll
    for (int n = 0; n < 8; ++n) {
      int c = n * 16 + l;
#pragma unroll
      for (int r = 0; r < 8; ++r) {
        int i = wave * 16 + r + 8 * hf;
        if (i < T_)
          atomicAdd(&out[((size_t)b2 * T_ + i) * C_ + c], acco[n][r]);
      }
    }
  }
}

// ---------------------------------------------------------------------------
extern "C" void kernel_launch(void* const* d_in, const int* in_sizes, int n_in,
                              void* d_out, int out_size, void* d_ws, size_t ws_size,
                              hipStream_t stream) {
  (void)in_sizes; (void)n_in; (void)out_size; (void)ws_size;
  const float* x      = (const float*)d_in[0];
  const float* Aadj   = (const float*)d_in[1];
  const float* conv_w = (const float*)d_in[2];
  const float* conv_b = (const float*)d_in[3];
  const float* bng    = (const float*)d_in[4];
  const float* bnb    = (const float*)d_in[5];
  const float* bnm    = (const float*)d_in[6];
  const float* bnv    = (const float*)d_in[7];
  const float* down_w = (const float*)d_in[8];
  const float* down_b = (const float*)d_in[9];
  const float* dbg    = (const float*)d_in[10];
  const float* dbb    = (const float*)d_in[11];
  const float* dbm    = (const float*)d_in[12];
  const float* dbv    = (const float*)d_in[13];
  const float* out_w  = (const float*)d_in[14];
  const float* out_b  = (const float*)d_in[15];

  float* out  = (float*)d_out;                         // (B2,T,C)
  float* attn = out + (size_t)B2_ * T_ * C_;           // (B2,H,T,T)

  float* qkv   = (float*)d_ws;                         // 73,728,000 f32
  float* Wfold = qkv + (size_t)B2_ * T_ * QKV_C;       // 512*1536 f32
  float* biasv = Wfold + (size_t)KCAT * QKV_C;         // 1536 f32

  fold_weights<<<(KCAT * QKV_C + 255) / 256, 256, 0, stream>>>(
      conv_w, conv_b, bng, bnb, bnm, bnv, down_w, down_b, dbg, dbb, dbm, dbv,
      Wfold, biasv);

  init_out<<<(B2_ * T_ * C_ + 255) / 256, 256, 0, stream>>>(out_b, out);

  size_t lds1 = (size_t)(TB * 32 * (KCAT + 4) + NHA * V_ * V_) * sizeof(float);
  gcn_qkv<<<B_ * (T_ / TB), 256, lds1, stream>>>(x, Aadj, Wfold, biasv, qkv);

  size_t lds2 = (size_t)(128 * 68 * 4 + 128 * 132) * sizeof(float);
  attention<<<dim3(B2_, HEADS), 256, lds2, stream>>>(qkv, out_w, out, attn);
}